// FastRCNNPredictor_24163486007539
// MI455X (gfx1250) — compile-verified
//
#include <hip/hip_runtime.h>
#include <hip/hip_bf16.h>
#include <math.h>

// ---------------------------------------------------------------------------
// FastRCNNPredictor eval path for MI455X (gfx1250, wave32, WMMA).
//   P   = avgpool7x7(x)                      [2048,2048]  (HBM-bound ~822MB, NT loads)
//   MAT = (P P^T) / max(n_i n_j, eps)        fp32 WMMA GEMM + cosine epilogue
//   Y   = MAT @ P ; softmax rows             fp32 WMMA GEMM
//   X1  = Y @ Wt^T + bt + P ; repeat w/ Wr -> X2 ; heads -> cls/bbox
// GEMM: V_WMMA_F32_16X16X4_F32. B tile staged in LDS (shared by the block's
// 4 waves, padded stride 36 -> conflict-free ds_load_b64 fragments). Inner
// loop branch-free; out-of-range columns handled by address clamping.
// ---------------------------------------------------------------------------

typedef float v2f __attribute__((ext_vector_type(2)));
typedef float v8f __attribute__((ext_vector_type(8)));

#define EPI_NONE     0
#define EPI_COS      1
#define EPI_BIAS_RES 2
#define EPI_BIAS     3

#define NN 2048
#define DD 2048

#define KC        32          // K chunk staged per barrier
#define BS_STRIDE 36          // padded LDS row stride (floats): 16B-aligned, 64-bank spread

// --------------------------- avgpool 7x7 -----------------------------------
__global__ __launch_bounds__(256)
void avgpool49_kernel(const float* __restrict__ x, float* __restrict__ P, int total)
{
    int wave = blockIdx.x * (blockDim.x >> 5) + (threadIdx.x >> 5);
    int lane = threadIdx.x & 31;
    if (wave >= total) return;
    size_t base = (size_t)wave * 49u;
    float s  = __builtin_nontemporal_load(&x[base + lane]);
    float s2 = __builtin_nontemporal_load(&x[base + 32 + (lane & 15)]);
    if (lane < 17) s += s2;
    #pragma unroll
    for (int off = 16; off; off >>= 1) s += __shfl_xor(s, off, 32);
    if (lane == 0) P[wave] = s * (1.0f / 49.0f);
}

// --------------------------- row L2 norms ----------------------------------
__global__ __launch_bounds__(256)
void rownorm_kernel(const float* __restrict__ X, float* __restrict__ norms,
                    int rows, int cols)
{
    int row  = blockIdx.x * (blockDim.x >> 5) + (threadIdx.x >> 5);
    int lane = threadIdx.x & 31;
    if (row >= rows) return;
    const float* r = X + (size_t)row * cols;
    float s = 0.0f;
    for (int j = lane; j < cols; j += 32) { float v = r[j]; s += v * v; }
    #pragma unroll
    for (int off = 16; off; off >>= 1) s += __shfl_xor(s, off, 32);
    if (lane == 0) norms[row] = sqrtf(s);
}

// --------------------------- row softmax (in place) ------------------------
__global__ __launch_bounds__(256)
void softmax_rows_kernel(float* __restrict__ Y, int cols)
{
    __shared__ float red[256];
    int tid = threadIdx.x;
    float* r = Y + (size_t)blockIdx.x * cols;

    float m = -INFINITY;
    for (int j = tid; j < cols; j += 256) m = fmaxf(m, r[j]);
    red[tid] = m; __syncthreads();
    for (int s = 128; s > 0; s >>= 1) {
        if (tid < s) red[tid] = fmaxf(red[tid], red[tid + s]);
        __syncthreads();
    }
    m = red[0]; __syncthreads();

    float sum = 0.0f;
    for (int j = tid; j < cols; j += 256) {
        float e = __expf(r[j] - m);
        r[j] = e; sum += e;
    }
    red[tid] = sum; __syncthreads();
    for (int s = 128; s > 0; s >>= 1) {
        if (tid < s) red[tid] += red[tid + s];
        __syncthreads();
    }
    float inv = 1.0f / red[0];
    for (int j = tid; j < cols; j += 256) r[j] *= inv;
}

// --------------------------- fp32 WMMA GEMM (LDS-staged B) -----------------
// C[M,Nn] = epilogue( A[M,K] * op(B) ), V_WMMA_F32_16X16X4_F32.
// TRANSB=1: op(B)[k][n] = B[n*ldb + k]   TRANSB=0: op(B)[k][n] = B[k*ldb + n]
// Block = 128 threads (4 waves). Block tile: 64 rows x 64 cols. The 64-wide
// B tile is staged through LDS in KC=32 chunks, shared by all 4 waves.
// LDS layout Bs[n][kk], stride 36 floats: fragment ds_load_b64 at
// n*36 + ka hits all 64 banks conflict-free.
template<int TRANSB, int EPI>
__global__ __launch_bounds__(128)
void gemm_wmma(const float* __restrict__ A, int lda,
               const float* __restrict__ B, int ldb,
               float* __restrict__ C, int ldc,
               int M, int Nn, int K,
               const float* __restrict__ bias,
               const float* __restrict__ resid, int ldr,
               const float* __restrict__ normI,
               const float* __restrict__ normJ)
{
    __shared__ float Bs[64 * BS_STRIDE];          // 9216 B

    const int tid  = threadIdx.x;
    const int wave = tid >> 5;
    const int lane = tid & 31;
    const int lrow = lane & 15;
    const int half = lane >> 4;

    const int m0     = blockIdx.y * 64 + wave * 16;
    const int n_base = blockIdx.x * 64;

    v8f acc[4] = {};
    const float* __restrict__ Arow = A + (size_t)(m0 + lrow) * lda;

    // Cooperative staging addresses (loop-invariant).
    const float* Bg;          // this thread's global source (at k0 = 0)
    int n_s = 0, ks = 0, kk_s = 0, ns = 0;
    if (TRANSB) {
        n_s = tid >> 1;                 // 0..63 : column of the tile
        ks  = (tid & 1) * 16;           // 0 | 16 : k offset within chunk
        int nc = n_base + n_s;
        nc = nc < (Nn - 1) ? nc : (Nn - 1);       // clamp (cols never stored)
        Bg = B + (size_t)nc * ldb + ks;           // 16 consecutive k per thread
    } else {
        kk_s = tid >> 2;                // 0..31 : k row within chunk
        ns   = (tid & 3) * 16;          // 0..48 : n offset (16 consecutive n)
        Bg = B + (size_t)kk_s * ldb + n_base + ns; // coalesced row loads
    }

    for (int k0 = 0; k0 < K; k0 += KC) {
        __syncthreads();
        if (TRANSB) {
            const float* src = Bg + k0;
            float* dst = &Bs[n_s * BS_STRIDE + ks];
            #pragma unroll
            for (int i = 0; i < 4; ++i)
                *(float4*)(dst + 4 * i) = *(const float4*)(src + 4 * i);
        } else {
            const float* src = Bg + (size_t)k0 * ldb;
            #pragma unroll
            for (int i = 0; i < 4; ++i) {
                float4 v = *(const float4*)(src + 4 * i);
                Bs[(ns + 4 * i + 0) * BS_STRIDE + kk_s] = v.x;
                Bs[(ns + 4 * i + 1) * BS_STRIDE + kk_s] = v.y;
                Bs[(ns + 4 * i + 2) * BS_STRIDE + kk_s] = v.z;
                Bs[(ns + 4 * i + 3) * BS_STRIDE + kk_s] = v.w;
            }
        }
        __syncthreads();

        #pragma unroll
        for (int kk = 0; kk < KC; kk += 4) {
            const int ka = kk + 2 * half;          // frag layout: v0 = K 0|2, v1 = K 1|3
            v2f a = *(const v2f*)(Arow + k0 + ka); // global_load_b64 (row-contiguous)
            #pragma unroll
            for (int t = 0; t < 4; ++t) {
                v2f b = *(const v2f*)(&Bs[(t * 16 + lrow) * BS_STRIDE + ka]); // ds_load_b64
                acc[t] = __builtin_amdgcn_wmma_f32_16x16x4_f32(
                    false, a, false, b, (short)0, acc[t], false, false);
            }
        }
    }

    // Epilogue. C/D layout: VGPR r -> M = m0 + r + 8*half, N = n_base + t*16 + lrow.
    #pragma unroll
    for (int t = 0; t < 4; ++t) {
        const int nc = n_base + t * 16 + lrow;
        if (nc < Nn) {
            const float bi = (EPI == EPI_BIAS_RES || EPI == EPI_BIAS) ? bias[nc] : 0.0f;
            const float nj = (EPI == EPI_COS) ? normJ[nc] : 0.0f;
            #pragma unroll
            for (int r = 0; r < 8; ++r) {
                const int m = m0 + r + 8 * half;
                float v = acc[t][r];
                if (EPI == EPI_COS)           v = v / fmaxf(normI[m] * nj, 1e-8f);
                else if (EPI == EPI_BIAS_RES) v += bi + resid[(size_t)m * ldr + nc];
                else if (EPI == EPI_BIAS)     v += bi;
                C[(size_t)m * ldc + nc] = v;
            }
        }
    }
}

// --------------------------- launch helpers --------------------------------
template<int TRANSB, int EPI>
static inline void launch_gemm(const float* A, int lda, const float* B, int ldb,
                               float* C, int ldc, int M, int Nn, int K,
                               const float* bias, const float* resid, int ldr,
                               const float* nI, const float* nJ,
                               hipStream_t stream)
{
    dim3 grid((Nn + 63) / 64, M / 64);
    gemm_wmma<TRANSB, EPI><<<grid, 128, 0, stream>>>(A, lda, B, ldb, C, ldc,
                                                     M, Nn, K, bias, resid, ldr, nI, nJ);
}

extern "C" void kernel_launch(void* const* d_in, const int* in_sizes, int n_in,
                              void* d_out, int out_size, void* d_ws, size_t ws_size,
                              hipStream_t stream)
{
    const float* x  = (const float*)d_in[0];   // [2048,2048,7,7]
    const float* Wt = (const float*)d_in[1];   // [2048,2048]
    const float* bt = (const float*)d_in[2];   // [2048]
    const float* Wr = (const float*)d_in[3];   // [2048,2048]
    const float* br = (const float*)d_in[4];   // [2048]
    const float* Wc = (const float*)d_in[5];   // [9,2048]
    const float* bc = (const float*)d_in[6];   // [9]
    const float* Wb = (const float*)d_in[7];   // [36,2048]
    const float* bb = (const float*)d_in[8];   // [36]
    // d_in[9] domain_masks, d_in[10] labels: unused on eval path.

    float* out = (float*)d_out;                // cls [2048,9] then bbox [2048,36]

    const size_t NE = (size_t)NN * DD;
    float* B0     = (float*)d_ws;              // P -> MAT2 -> X2
    float* B1     = B0 + NE;                   // MAT -> X1
    float* B2     = B1 + NE;                   // Y (softmax operand)
    float* norms1 = B2 + NE;
    float* norms2 = norms1 + NN;

    // 1) P = avgpool(x)
    {
        int total  = NN * DD;                  // one wave per output
        int blocks = total / 8;                // 8 waves / 256-thread block
        avgpool49_kernel<<<blocks, 256, 0, stream>>>(x, B0, total);
    }
    // 2) norms of P
    rownorm_kernel<<<NN / 8, 256, 0, stream>>>(B0, norms1, NN, DD);
    // 3) MAT = cos-sim(P)
    launch_gemm<1, EPI_COS>(B0, DD, B0, DD, B1, NN, NN, NN, DD,
                            nullptr, nullptr, 0, norms1, norms1, stream);
    // 4) Y = MAT @ P
    launch_gemm<0, EPI_NONE>(B1, NN, B0, DD, B2, DD, NN, DD, NN,
                             nullptr, nullptr, 0, nullptr, nullptr, stream);
    // 5) softmax rows of Y
    softmax_rows_kernel<<<NN, 256, 0, stream>>>(B2, DD);
    // 6) X1 = Y @ Wt^T + bt + P
    launch_gemm<1, EPI_BIAS_RES>(B2, DD, Wt, DD, B1, DD, NN, DD, DD,
                                 bt, B0, DD, nullptr, nullptr, stream);
    // 7) norms of X1
    rownorm_kernel<<<NN / 8, 256, 0, stream>>>(B1, norms2, NN, DD);
    // 8) MAT2 = cos-sim(X1)
    launch_gemm<1, EPI_COS>(B1, DD, B1, DD, B0, NN, NN, NN, DD,
                            nullptr, nullptr, 0, norms2, norms2, stream);
    // 9) Y2 = MAT2 @ X1
    launch_gemm<0, EPI_NONE>(B0, NN, B1, DD, B2, DD, NN, DD, NN,
                             nullptr, nullptr, 0, nullptr, nullptr, stream);
    // 10) softmax rows of Y2
    softmax_rows_kernel<<<NN, 256, 0, stream>>>(B2, DD);
    // 11) X2 = Y2 @ Wr^T + br + X1
    launch_gemm<1, EPI_BIAS_RES>(B2, DD, Wr, DD, B0, DD, NN, DD, DD,
                                 br, B1, DD, nullptr, nullptr, stream);
    // 12) cls = X2 @ Wc^T + bc
    launch_gemm<1, EPI_BIAS>(B0, DD, Wc, DD, out, 9, NN, 9, DD,
                             bc, nullptr, 0, nullptr, nullptr, stream);
    // 13) bbox = X2 @ Wb^T + bb
    launch_gemm<1, EPI_BIAS>(B0, DD, Wb, DD, out + (size_t)NN * 9, 36, NN, 36, DD,
                             bb, nullptr, 0, nullptr, nullptr, stream);
}